// DownsampleBlock_52716428591193
// MI455X (gfx1250) — compile-verified
//
#include <hip/hip_runtime.h>
#include <hip/hip_bf16.h>

typedef __attribute__((ext_vector_type(2))) float v2f;
typedef __attribute__((ext_vector_type(4))) float v4f;
typedef __attribute__((ext_vector_type(8))) float v8f;
typedef int v4i __attribute__((vector_size(16)));
typedef __attribute__((address_space(1))) v4i g_v4i;   // global-AS 16B vector
typedef __attribute__((address_space(3))) v4i l_v4i;   // LDS-AS 16B vector

#define HW    160        // input H = W
#define OWD   80         // output H = W
#define CIN   64
#define COUT  128
#define KTOT  256        // CIN*2*2
#define NTOT  (32*OWD*OWD)   // 204800
// LDS Wq layout: [kpair 0..127][cout 0..127][2], pitch 288 dwords per kpair row.
// 288 % 64 == 32 -> kh=0 half-wave uses banks {c..c+31}, kh=1 half-wave the
// other 32 banks: conflict-free ds_load_b64 B fragments.
#define WQ_PITCH  288
#define OUT_PITCH 132    // dwords; 16B-aligned rows for b128 copy-out
#define SMEM_BYTES (128 * WQ_PITCH * 4)   // 147456 B (out tile 128*132*4 reuses it)

// ---------------------------------------------------------------------------
// Kernel 1: per-cout weight fake-quant; store Wq K-pair-interleaved in ws:
//   ws[(k>>1)*256 + cout*2 + (k&1)]   (k = cin*4 + kh*2 + kw)
// One wave per cout; each lane owns 8 of the 256 weights.
// ---------------------------------------------------------------------------
__global__ void quant_weights_kernel(const float* __restrict__ W,
                                     float* __restrict__ wq_t) {
  const int wave = threadIdx.x >> 5;
  const int lane = threadIdx.x & 31;
  const int cout = blockIdx.x * 8 + wave;          // grid = 16 blocks * 8 waves
  const float* wrow = W + cout * KTOT;

  float v[8];
  float m = 0.0f;
#pragma unroll
  for (int i = 0; i < 8; ++i) {
    v[i] = wrow[lane * 8 + i];
    m = fmaxf(m, fabsf(v[i]));
  }
#pragma unroll
  for (int off = 16; off >= 1; off >>= 1)
    m = fmaxf(m, __shfl_xor(m, off, 32));

  const float ws  = fmaxf(m * (1.0f / 63.0f), 1e-8f);
  const float inv = 1.0f / ws;
#pragma unroll
  for (int i = 0; i < 8; ++i) {
    float q = rintf(v[i] * inv);
    q = fminf(fmaxf(q, -64.0f), 63.0f);
    int k = lane * 8 + i;
    wq_t[(k >> 1) * 256 + cout * 2 + (k & 1)] = q * ws;
  }
}

// ---------------------------------------------------------------------------
// Kernel 2: conv-as-GEMM with fp32 WMMA 16x16x4.
//   A (16n x 4k) = x patches, per-lane contiguous float2 (kw pair), hi = kh.
//   B (4k x 16cout) = one ds_load_b64 per fragment (pair-interleaved LDS).
//   D (16n x 16cout) x 8 cout-tiles per wave.
// ---------------------------------------------------------------------------
extern __shared__ float smem[];

__global__ void __launch_bounds__(256)
conv_q_wmma_kernel(const float* __restrict__ x,
                   const float* __restrict__ wq_t,
                   const float* __restrict__ bias,
                   const float* __restrict__ a_scale,
                   const float* __restrict__ a_zp,
                   float* __restrict__ out) {
  const int t = threadIdx.x;

  // ---- Stage Wq (128 KB) -> LDS, b128 granules; async-to-LDS if available ----
  {
#if __has_builtin(__builtin_amdgcn_global_load_async_to_lds_b128)
    auto gsrc  = (__attribute__((address_space(1))) char*)wq_t;
    auto lbase = (__attribute__((address_space(3))) char*)smem;
#pragma unroll
    for (int i = 0; i < 32; ++i) {
      int j   = t + 256 * i;                       // 0..8191 float4s
      int row = j >> 6;                            // kpair row (64 float4 / row)
      int c4  = j & 63;
      __builtin_amdgcn_global_load_async_to_lds_b128(
          (g_v4i*)(gsrc + (size_t)j * 16),
          (l_v4i*)(lbase + (size_t)(row * WQ_PITCH + c4 * 4) * 4),
          0, 0);
    }
#if __has_builtin(__builtin_amdgcn_s_wait_asynccnt)
    __builtin_amdgcn_s_wait_asynccnt(0);
#else
    asm volatile("s_wait_asynccnt 0x0" ::: "memory");
#endif
#else
    const v4f* src = (const v4f*)wq_t;             // [128][64] float4
#pragma unroll
    for (int i = 0; i < 32; ++i) {
      int j   = t + 256 * i;
      int row = j >> 6;
      int c4  = j & 63;
      v4f val = src[row * 64 + c4];
      *(v4f*)&smem[row * WQ_PITCH + c4 * 4] = val;
    }
#endif
  }
  __syncthreads();

  const int wave = t >> 5;
  const int lane = t & 31;
  const int nl   = lane & 15;                      // n (A rows) / cout (B cols)
  const int hi   = lane >> 4;                      // K pair select (== kh)

  const int n0 = blockIdx.x * 128 + wave * 16;     // wave's first output pixel
  const int b  = n0 / (OWD * OWD);
  const int sp = n0 % (OWD * OWD);
  const int oh = sp / OWD;
  const int ow0 = sp % OWD;                        // multiple of 16: no row straddle

  // A-frag pointer: x[b][cin=0][2*oh+hi][2*ow0 + 2*nl]; advances HW*HW per cin
  const float* xp = x + ((b * CIN) * HW + (2 * oh + hi)) * HW + 2 * ow0 + 2 * nl;

  v8f zero = {};
  v8f acc[8];
#pragma unroll
  for (int i = 0; i < 8; ++i) acc[i] = zero;

#pragma unroll 4
  for (int ks = 0; ks < 64; ++ks) {                // one cin per K-step of 4
    v2f a = *(const v2f*)xp;                       // coalesced 128B per half-wave
    xp += HW * HW;
    // kpair row = (ks*4 + 2*hi)/2 = ks*2 + hi ; column nl, pair contiguous
    const float* wrow = &smem[(ks * 2 + hi) * WQ_PITCH + nl * 2];
#pragma unroll
    for (int ct = 0; ct < 8; ++ct) {               // 8 cout tiles of 16
      v2f bf = *(const v2f*)&wrow[ct * 32];        // single ds_load_b64
      acc[ct] = __builtin_amdgcn_wmma_f32_16x16x4_f32(
          false, a, false, bf, (short)0, acc[ct], false, false);
    }
  }

  // ---- Epilogue: bias + activation fake-quant in registers ----
  float bs[8], sc[8], iv[8], zp[8];
#pragma unroll
  for (int ct = 0; ct < 8; ++ct) {
    int c = ct * 16 + nl;
    bs[ct] = bias[c];
    sc[ct] = a_scale[c];
    zp[ct] = a_zp[c];
    iv[ct] = 1.0f / sc[ct];
  }

  __syncthreads();                                 // Wq done; reuse LDS for out tile
#pragma unroll
  for (int ct = 0; ct < 8; ++ct) {
    int cout = ct * 16 + nl;
#pragma unroll
    for (int r = 0; r < 8; ++r) {
      float y = acc[ct][r] + bs[ct];
      float q = rintf(y * iv[ct]) + zp[ct];
      q = fminf(fmaxf(q, -64.0f), 63.0f);
      float o = (q - zp[ct]) * sc[ct];
      int nloc = wave * 16 + r + 8 * hi;           // D row = r + 8*hi
      smem[cout * OUT_PITCH + nloc] = o;
    }
  }
  __syncthreads();

  // ---- Coalesced b128 copy-out: [128 cout][128 n] tile ----
  const int bb  = (blockIdx.x * 128) / (OWD * OWD);
  const int spw = (blockIdx.x * 128) % (OWD * OWD); // multiple of 128 (6400/128=50)
#pragma unroll
  for (int i = 0; i < 16; ++i) {
    int j    = t + 256 * i;                        // 0..4095
    int cout = j >> 5;                             // 32 float4 per cout row
    int n4   = (j & 31) * 4;
    v4f val  = *(const v4f*)&smem[cout * OUT_PITCH + n4];
    *(v4f*)&out[(size_t)(bb * COUT + cout) * (OWD * OWD) + spw + n4] = val;
  }
}

// ---------------------------------------------------------------------------
extern "C" void kernel_launch(void* const* d_in, const int* in_sizes, int n_in,
                              void* d_out, int out_size, void* d_ws, size_t ws_size,
                              hipStream_t stream) {
  const float* x       = (const float*)d_in[0];
  const float* W       = (const float*)d_in[1];
  const float* bias    = (const float*)d_in[2];
  const float* a_scale = (const float*)d_in[3];
  const float* a_zp    = (const float*)d_in[4];
  float* out = (float*)d_out;
  float* wq_t = (float*)d_ws;                      // 32768 floats = 128 KB

  quant_weights_kernel<<<16, 256, 0, stream>>>(W, wq_t);

  const int nblocks = NTOT / 128;                  // 1600
  conv_q_wmma_kernel<<<nblocks, 256, SMEM_BYTES, stream>>>(
      x, wq_t, bias, a_scale, a_zp, out);
}